// SamplingAndGrouping_68195490726012
// MI455X (gfx1250) — compile-verified
//
#include <hip/hip_runtime.h>
#include <hip/hip_bf16.h>

// ---------------------------------------------------------------------------
// PointNet++ sampling & grouping for MI455X (gfx1250, wave32).
//   FPS       : 1 workgroup / batch; whole 240KB point cloud resident in the
//               CDNA5 320KB LDS (SoA x/y/z); packed-u64 argmax reduction.
//   BallQuery : 2x V_WMMA_F32_16X16X4_F32 per 32 candidates:
//               d2 = (-2 q . c + |q|^2) + |c|^2  (|c|^2 folded into C init),
//               coalesced candidate loads (1 distinct candidate per lane),
//               wave32 ballot scan picks the 64 smallest passing indices.
//   Grouping  : flat gather (bandwidth-trivial at 23.3 TB/s).
// ---------------------------------------------------------------------------

#define BATCH   4
#define NPTS    20000
#define NPOINT  2048
#define NSAMPLE 64
#define NFEAT   6
#define RAD2    0.04f

typedef float v2f __attribute__((ext_vector_type(2)));
typedef float v8f __attribute__((ext_vector_type(8)));

// d_out flat offsets (reference tuple return order)
#define OUT_NEWXYZ   0                                   // B*S*3   = 24576
#define OUT_INDS     (BATCH * NPOINT * 3)                // B*S     = 8192
#define OUT_GROUPED  (OUT_INDS + BATCH * NPOINT)         // B*S*K*9 = 4718592
#define OUT_PAINTED  (OUT_GROUPED + BATCH * NPOINT * NSAMPLE * 9)  // B*N = 80000

// ---------------------------------------------------------------------------
// Kernel 1: farthest point sampling. One 1024-thread workgroup per batch.
// The batch's 20000 points are staged once into LDS (3 x 80000B SoA = 240KB,
// possible only because CDNA5 gives 320KB LDS per workgroup), so each of the
// 2047 serial iterations runs off LDS latency instead of L2 latency.
// dists live in registers (20 per thread); argmax via packed u64 max:
//   key = (f32_bits(dist) << 32) | (0xFFFFFFFF - index)
// dist >= 0 so float bit pattern is order-preserving; ties -> smaller index,
// matching jnp.argmax first-max semantics.
// ---------------------------------------------------------------------------
#define FPS_T 1024
#define FPS_K 20   // ceil(20000/1024)

__global__ __launch_bounds__(FPS_T)
void fps_kernel(const float* __restrict__ xyz, int* __restrict__ inds_ws)
{
    __shared__ float lx[NPTS];                 // 80000 B
    __shared__ float ly[NPTS];                 // 80000 B
    __shared__ float lz[NPTS];                 // 80000 B
    __shared__ unsigned long long red[32];     // 256 B   (total ~240.3 KB)

    const int b = blockIdx.x;
    const int t = threadIdx.x;
    const float* xb = xyz + (size_t)b * NPTS * 3;

    // stage point cloud into LDS once
#pragma unroll
    for (int k = 0; k < FPS_K; ++k) {
        const int j = t + k * FPS_T;
        if (j < NPTS) {
            lx[j] = xb[j * 3 + 0];
            ly[j] = xb[j * 3 + 1];
            lz[j] = xb[j * 3 + 2];
        }
    }

    float dist[FPS_K];
#pragma unroll
    for (int k = 0; k < FPS_K; ++k) dist[k] = 1.0e10f;

    int last = 0;
    if (t == 0) inds_ws[b * NPOINT] = 0;
    __syncthreads();

    for (int it = 1; it < NPOINT; ++it) {
        const float px = lx[last];
        const float py = ly[last];
        const float pz = lz[last];

        unsigned long long best = 0ull;
#pragma unroll
        for (int k = 0; k < FPS_K; ++k) {
            const int j = t + k * FPS_T;
            if (j < NPTS) {
                const float dx = lx[j] - px;
                const float dy = ly[j] - py;
                const float dz = lz[j] - pz;
                const float d  = dx * dx + dy * dy + dz * dz;
                const float nd = fminf(dist[k], d);
                dist[k] = nd;
                const unsigned long long key =
                    ((unsigned long long)__float_as_uint(nd) << 32) |
                    (unsigned long long)(0xFFFFFFFFu - (unsigned)j);
                best = (key > best) ? key : best;
            }
        }
        // wave32 reduction
#pragma unroll
        for (int off = 16; off > 0; off >>= 1) {
            const unsigned long long o = __shfl_xor(best, off, 32);
            best = (o > best) ? o : best;
        }
        if ((t & 31) == 0) red[t >> 5] = best;
        __syncthreads();
        if (t < 32) {
            unsigned long long v = red[t];
#pragma unroll
            for (int off = 16; off > 0; off >>= 1) {
                const unsigned long long o = __shfl_xor(v, off, 32);
                v = (o > v) ? o : v;
            }
            if (t == 0) red[0] = v;
        }
        __syncthreads();
        const unsigned long long win = red[0];
        __syncthreads();   // protect red[] before next iteration's writes
        last = (int)(0xFFFFFFFFu - (unsigned)(win & 0xFFFFFFFFull));
        if (t == 0) inds_ws[b * NPOINT + it] = last;
    }
}

// ---------------------------------------------------------------------------
// Kernel 2: gather new_xyz into d_out, emit inds as float.
// ---------------------------------------------------------------------------
__global__ void gather_newxyz_kernel(const float* __restrict__ xyz,
                                     const int* __restrict__ inds_ws,
                                     float* __restrict__ out)
{
    const int tid = blockIdx.x * blockDim.x + threadIdx.x;
    if (tid >= BATCH * NPOINT) return;
    const int b = tid / NPOINT;
    const int j = inds_ws[tid];
    const float* p = xyz + ((size_t)b * NPTS + j) * 3;
    float* nx = out + OUT_NEWXYZ + (size_t)tid * 3;
    nx[0] = p[0]; nx[1] = p[1]; nx[2] = p[2];
    out[OUT_INDS + tid] = (float)j;
}

// ---------------------------------------------------------------------------
// Kernel 3: ball query via V_WMMA_F32_16X16X4_F32, 32 candidates / iteration.
// One wave per 16-query tile. Each lane loads ONE distinct candidate
// (coalesced b96); __shfl redistributes cz / |c|^2 into the two B fragments.
//   A  (16x4 f32): lanes 0-15 (K0,K1)=(-2qx,-2qy) of row m=lane,
//                  lanes 16-31 (K2,K3)=(-2qz, |q|^2).
//   B0 (4x16 f32): cols = candidates t0..t0+15
//   B1 (4x16 f32): cols = candidates t0+16..t0+31
//   C init: per-lane |c_n|^2 (C/D columns map to lane&15).
//   D = -2 q.c + |q|^2 + |c|^2 = squared distance.
// Selection: lane m<16 owns query row m; combined 32-bit mask has tile0 in
// bits 0-15 and tile1 in bits 16-31, so ascending bit position == ascending
// candidate index (j = t0 + ctz). Uniform early exit keeps EXEC all-ones.
// ---------------------------------------------------------------------------
__global__ __launch_bounds__(256)
void ballquery_wmma_kernel(const float* __restrict__ xyz,
                           const float* __restrict__ out_newxyz,
                           int* __restrict__ idx_ws)
{
    const int gtid  = blockIdx.x * blockDim.x + threadIdx.x;
    const int wave  = gtid >> 5;
    const int lane  = threadIdx.x & 31;
    const int tilesPerBatch = NPOINT / 16;            // 128
    const int b    = wave / tilesPerBatch;            // grid sized exactly
    const int tile = wave % tilesPerBatch;
    const int s0   = tile * 16;
    const int m    = lane & 15;

    // A-matrix fragment for this wave's 16 queries
    const float* q = out_newxyz + ((size_t)b * NPOINT + s0 + m) * 3;
    const float qx = q[0], qy = q[1], qz = q[2];
    const float qn = qx * qx + qy * qy + qz * qz;
    v2f A;
    if (lane < 16) { A.x = -2.0f * qx; A.y = -2.0f * qy; }
    else           { A.x = -2.0f * qz; A.y = qn;         }

    const float* xb = xyz + (size_t)b * NPTS * 3;
    int cnt = 0, first = -1;
    const int outbase = ((b * NPOINT) + s0 + m) * NSAMPLE;

    for (int t0 = 0; t0 < NPTS; t0 += 32) {
        // each lane loads one distinct candidate: j = t0 + lane
        const float* c = xb + (size_t)(t0 + lane) * 3;
        const float cx = c[0], cy = c[1], cz = c[2];
        const float cn2 = cx * cx + cy * cy + cz * cz;

        // redistribute for the two B fragments
        const float czlo  = __shfl(cz, m, 32);        // cz of cand t0+(lane-16)
        const float cxhi  = __shfl(cx, m + 16, 32);   // cx of cand t0+16+lane
        const float cyhi  = __shfl(cy, m + 16, 32);
        const float cn2_0 = __shfl(cn2, m, 32);       // |c|^2, col n = lane&15
        const float cn2_1 = __shfl(cn2, m + 16, 32);

        v2f B0, B1;
        if (lane < 16) { B0.x = cx;   B0.y = cy;   B1.x = cxhi; B1.y = cyhi; }
        else           { B0.x = czlo; B0.y = 1.0f; B1.x = cz;   B1.y = 1.0f; }

        v8f C0, C1;
#pragma unroll
        for (int r = 0; r < 8; ++r) { C0[r] = cn2_0; C1[r] = cn2_1; }

        v8f D0 = __builtin_amdgcn_wmma_f32_16x16x4_f32(
            false, A, false, B0, (short)0, C0, false, false);
        v8f D1 = __builtin_amdgcn_wmma_f32_16x16x4_f32(
            false, A, false, B1, (short)0, C1, false, false);

        // Per-row pass masks: reg r -> rows r (bits 15:0) and r+8 (bits 31:16)
        unsigned mymask = 0u;
#pragma unroll
        for (int r = 0; r < 8; ++r) {
            const unsigned f0 = (unsigned)__ballot(D0[r] < RAD2);
            const unsigned f1 = (unsigned)__ballot(D1[r] < RAD2);
            if ((m & 7) == r) {
                const unsigned lo = ((m >> 3) ? (f0 >> 16) : f0) & 0xFFFFu;
                const unsigned hi = ((m >> 3) ? (f1 >> 16) : f1) & 0xFFFFu;
                mymask = lo | (hi << 16);
            }
        }
        if (lane >= 16) mymask = 0u;

        while (mymask && cnt < NSAMPLE) {
            const int bit = __builtin_ctz(mymask);
            const int j   = t0 + bit;
            if (first < 0) first = j;
            idx_ws[outbase + cnt] = j;
            ++cnt;
            mymask &= mymask - 1u;
        }

        // uniform early exit keeps EXEC all-ones for the next WMMA
        if (__ballot((lane < 16) && (cnt < NSAMPLE)) == 0ull) break;
    }

    if (lane < 16) {
        const int pad = (first < 0) ? 0 : first;
        for (int k = cnt; k < NSAMPLE; ++k) idx_ws[outbase + k] = pad;
    }
}

// ---------------------------------------------------------------------------
// Kernel 4: grouping gather — grouped_xyz(3) | grouped_feat(6) per (b,s,k).
// ---------------------------------------------------------------------------
__global__ __launch_bounds__(256)
void group_kernel(const float* __restrict__ xyz,
                  const float* __restrict__ feat,
                  const float* __restrict__ out_newxyz,
                  const int* __restrict__ idx_ws,
                  float* __restrict__ out)
{
    const int tid = blockIdx.x * blockDim.x + threadIdx.x;
    if (tid >= BATCH * NPOINT * NSAMPLE) return;
    const int bs = tid / NSAMPLE;          // b*NPOINT + s
    const int b  = bs / NPOINT;
    const int j  = idx_ws[tid];

    const float* p = xyz + ((size_t)b * NPTS + j) * 3;
    const float* q = out_newxyz + (size_t)bs * 3;
    const float* f = feat + ((size_t)b * NPTS + j) * NFEAT;
    float* o = out + OUT_GROUPED + (size_t)tid * (3 + NFEAT);

    o[0] = p[0] - q[0];
    o[1] = p[1] - q[1];
    o[2] = p[2] - q[2];
#pragma unroll
    for (int c = 0; c < NFEAT; ++c) o[3 + c] = f[c];
}

// ---------------------------------------------------------------------------
// Kernel 5: isPainted passthrough (int -> float slot in d_out).
// ---------------------------------------------------------------------------
__global__ void painted_kernel(const int* __restrict__ isPainted,
                               float* __restrict__ out)
{
    const int tid = blockIdx.x * blockDim.x + threadIdx.x;
    if (tid >= BATCH * NPTS) return;
    out[OUT_PAINTED + tid] = (float)isPainted[tid];
}

// ---------------------------------------------------------------------------
extern "C" void kernel_launch(void* const* d_in, const int* in_sizes, int n_in,
                              void* d_out, int out_size, void* d_ws, size_t ws_size,
                              hipStream_t stream)
{
    const float* xyz       = (const float*)d_in[0];   // (4,20000,3) f32
    const int*   isPainted = (const int*)  d_in[1];   // (4,20000)   i32
    const float* features  = (const float*)d_in[2];   // (4,20000,6) f32
    float* out = (float*)d_out;

    // workspace: [0, B*S) ints = FPS inds ; then B*S*K ints = ball-query idx
    int* inds_ws = (int*)d_ws;
    int* idx_ws  = inds_ws + BATCH * NPOINT;

    // 1) FPS: one WGP-resident workgroup per batch (240KB LDS each)
    fps_kernel<<<BATCH, FPS_T, 0, stream>>>(xyz, inds_ws);

    // 2) new_xyz gather + inds->float
    gather_newxyz_kernel<<<(BATCH * NPOINT + 255) / 256, 256, 0, stream>>>(
        xyz, inds_ws, out);

    // 3) ball query: 512 waves total (4 batches * 128 query tiles), 8 waves/block
    {
        const int waves  = BATCH * (NPOINT / 16);     // 512
        const int blocks = waves * 32 / 256;          // 64, exact
        ballquery_wmma_kernel<<<blocks, 256, 0, stream>>>(
            xyz, out + OUT_NEWXYZ, idx_ws);
    }

    // 4) grouping gather
    group_kernel<<<(BATCH * NPOINT * NSAMPLE + 255) / 256, 256, 0, stream>>>(
        xyz, features, out + OUT_NEWXYZ, idx_ws, out);

    // 5) isPainted copy
    painted_kernel<<<(BATCH * NPTS + 255) / 256, 256, 0, stream>>>(isPainted, out);
}